// PSRoiAlignPooling_36567351558872
// MI455X (gfx1250) — compile-verified
//
#include <hip/hip_runtime.h>
#include <math.h>

// PS-RoI-Align (position-sensitive, bilinear, max-pooled) for gfx1250.
// Gather-latency-bound: image (80MB) is L2-resident (192MB L2). Optimize for
// (a) shortest address-generation dependency chain (no IEEE divides),
// (b) wide coalesced B128 gathers (4 lanes per bin cover a 64B pixel row),
// (c) 16 independent loads in flight per thread,
// (d) non-temporal B128 store for the write-only output (keeps L2 for img).

typedef float v4f __attribute__((ext_vector_type(4)));

namespace {

constexpr int POOL  = 7;
constexpr int BINS  = POOL * POOL;   // 49
constexpr int ALPHA = 16;
constexpr int Hdim  = 160;
constexpr int Wdim  = 160;
constexpr int CH    = BINS * ALPHA;  // 784 channels in img

__device__ __forceinline__ float lerpf(float a, float b, float t) {
  return a + (b - a) * t;            // matches reference: v0 + (v1-v0)*l
}

__global__ __launch_bounds__(256) void psroi_align_max_kernel(
    const float* __restrict__ img,   // [H, W, BINS*ALPHA]
    const float* __restrict__ rois,  // [R, 4] = xmin, ymin, xmax, ymax
    float* __restrict__ out,         // [R, BINS, ALPHA] (== R,POOL,POOL,ALPHA)
    int R)
{
  const int tid  = blockIdx.x * blockDim.x + threadIdx.x;
  const int pair = tid >> 2;                 // (r, k) bin index
  if (pair >= R * BINS) return;
  const int c4   = (tid & 3) * 4;            // which float4 of the 16 channels

  const int r  = pair / BINS;
  const int k  = pair - r * BINS;
  const int bx = k / POOL;                   // x-direction bin (k // POOL)
  const int by = k - bx * POOL;              // y-direction bin (k %  POOL)

  const v4f roi = *reinterpret_cast<const v4f*>(rois + 4 * r);
  // Multiply by reciprocal instead of IEEE divide: kills the v_div_scale/
  // v_div_fmas/v_div_fixup sequence (~20 serial VALU ops) on the address path.
  constexpr float INV_POOL = 1.0f / (float)POOL;
  const float stepx = (roi.z - roi.x) * INV_POOL;
  const float stepy = (roi.w - roi.y) * INV_POOL;

  const float xA = roi.x + (float)bx * stepx;
  const float yA = roi.y + (float)by * stepy;

  float sx[2] = { xA * (float)(Wdim - 1), (xA + stepx) * (float)(Wdim - 1) };
  float sy[2] = { yA * (float)(Hdim - 1), (yA + stepy) * (float)(Hdim - 1) };

  int   x0[2], x1[2], y0[2], y1[2];
  float lx[2], ly[2];
  bool  vx[2], vy[2];

#pragma unroll
  for (int j = 0; j < 2; ++j) {
    const float s = sx[j];
    vx[j] = (s >= 0.0f) && (s <= (float)(Wdim - 1));
    const float f = floorf(s);
    lx[j] = s - f;                                             // frac from UNclipped floor
    x0[j] = (int)fminf(fmaxf(f, 0.0f), (float)(Wdim - 1));
    x1[j] = (int)fminf(fmaxf(ceilf(s), 0.0f), (float)(Wdim - 1));
  }
#pragma unroll
  for (int i = 0; i < 2; ++i) {
    const float s = sy[i];
    vy[i] = (s >= 0.0f) && (s <= (float)(Hdim - 1));
    const float f = floorf(s);
    ly[i] = s - f;
    y0[i] = (int)fminf(fmaxf(f, 0.0f), (float)(Hdim - 1));
    y1[i] = (int)fminf(fmaxf(ceilf(s), 0.0f), (float)(Hdim - 1));
  }

  const int co = k * ALPHA + c4;             // channel offset inside a pixel

  v4f m = { -INFINITY, -INFINITY, -INFINITY, -INFINITY };

#pragma unroll
  for (int i = 0; i < 2; ++i) {
#pragma unroll
    for (int j = 0; j < 2; ++j) {
      // 4 independent B128 gathers per sample; fully unrolled so all 16
      // loads of the thread can be in flight before the first s_wait_loadcnt.
      const v4f v00 = *reinterpret_cast<const v4f*>(img + (size_t)(y0[i] * Wdim + x0[j]) * CH + co);
      const v4f v01 = *reinterpret_cast<const v4f*>(img + (size_t)(y0[i] * Wdim + x1[j]) * CH + co);
      const v4f v10 = *reinterpret_cast<const v4f*>(img + (size_t)(y1[i] * Wdim + x0[j]) * CH + co);
      const v4f v11 = *reinterpret_cast<const v4f*>(img + (size_t)(y1[i] * Wdim + x1[j]) * CH + co);
      const float wx = lx[j];
      const float wy = ly[i];
      const bool  ok = vx[j] && vy[i];
#define PSROI_COMP(c)                                                       \
      { const float t = lerpf(v00.c, v01.c, wx);                            \
        const float b = lerpf(v10.c, v11.c, wx);                            \
        const float v = lerpf(t, b, wy);                                    \
        m.c = fmaxf(m.c, ok ? v : 0.0f); }
      PSROI_COMP(x) PSROI_COMP(y) PSROI_COMP(z) PSROI_COMP(w)
#undef PSROI_COMP
    }
  }

  // Output is write-only and never re-read: non-temporal B128 store (TH=NT)
  // keeps L2 capacity dedicated to the (L2-resident) image.
  v4f* outp = reinterpret_cast<v4f*>(out + (size_t)pair * ALPHA + c4);
  __builtin_nontemporal_store(m, outp);
}

} // namespace

extern "C" void kernel_launch(void* const* d_in, const int* in_sizes, int n_in,
                              void* d_out, int out_size, void* d_ws, size_t ws_size,
                              hipStream_t stream) {
  const float* img  = (const float*)d_in[0];   // (1, 160, 160, 784) f32
  const float* rois = (const float*)d_in[1];   // (1, R, 4) f32
  float* out = (float*)d_out;                  // (1, R, 7, 7, 16) f32

  const int R = in_sizes[1] / 4;               // 512
  const int total_threads = R * BINS * 4;      // 4 lanes per (roi, bin)
  const int block = 256;
  const int grid  = (total_threads + block - 1) / block;

  psroi_align_max_kernel<<<grid, block, 0, stream>>>(img, rois, out, R);
}